// PERModel_18184891531632
// MI455X (gfx1250) — compile-verified
//
#include <hip/hip_runtime.h>
#include <math.h>

// ---------------------------------------------------------------------------
// Shapes
// ---------------------------------------------------------------------------
#define N_P   512
#define N_W   3584
#define N_WP  4096          // N_P + N_W
#define F_IN  256
#define H1    8
#define O1    32
#define H2    8
#define O2    256
#define BATCH 1024
#define NEG_SLOPE 0.2f

typedef float v2f __attribute__((ext_vector_type(2)));
typedef float v8f __attribute__((ext_vector_type(8)));
typedef int   v4i __attribute__((ext_vector_type(4)));

static __device__ __forceinline__ v8f vzero8() {
    v8f z = {0.f, 0.f, 0.f, 0.f, 0.f, 0.f, 0.f, 0.f};
    return z;
}

// D = A(16x4 f32) * B(4x16 f32) + C, full fp32 accumulate on the matrix core.
static __device__ __forceinline__ v8f wmma_f32_16x16x4(v2f a, v2f b, v8f c) {
    return __builtin_amdgcn_wmma_f32_16x16x4_f32(
        /*neg_a=*/false, a, /*neg_b=*/false, b,
        /*c_mod=*/(short)0, c, /*reuse_a=*/false, /*reuse_b=*/false);
}

// gfx1250 async global->LDS copy (ASYNCcnt-tracked, bypasses VGPRs).
// Builtin signature (from probe): (v4i addrspace(1)*, v4i addrspace(3)*, imm, imm)
static __device__ __forceinline__ void async_load_b128(const void* gptr, void* lptr) {
#if __has_builtin(__builtin_amdgcn_global_load_async_to_lds_b128)
    void* g = const_cast<void*>(gptr);
    __builtin_amdgcn_global_load_async_to_lds_b128(
        (__attribute__((address_space(1))) v4i*)g,
        (__attribute__((address_space(3))) v4i*)lptr, 0, 0);
#else
    unsigned loff = (unsigned)(__UINTPTR_TYPE__)(__attribute__((address_space(3))) void*)lptr;
    asm volatile("global_load_async_to_lds_b128 %0, %1, off"
                 :: "v"(loff), "v"(gptr) : "memory");
#endif
}

static __device__ __forceinline__ void wait_async0() {
#if __has_builtin(__builtin_amdgcn_s_wait_asynccnt)
    __builtin_amdgcn_s_wait_asynccnt(0);
#else
    asm volatile("s_wait_asynccnt 0x0" ::: "memory");
#endif
}

// ---------------------------------------------------------------------------
// GAT1 h_prime via WMMA: hp1[h][n][o] = pern_feature[n,:] . w1[h,:,o]
// Head dim folds into columns: col = h*32+o.  M=512 (32 tiles), N=256, K=256.
// grid 32, block 128 = 4 waves * 4 N-tiles
// ---------------------------------------------------------------------------
__global__ void k_gat1_hprime(const float* __restrict__ pf,
                              const float* __restrict__ w1,
                              float* __restrict__ hp1) {
    int mt = blockIdx.x;
    int lane = threadIdx.x & 31, wave = threadIdx.x >> 5;
    int hi = lane >> 4, l15 = lane & 15;
    const float* arow = pf + (size_t)(mt * 16 + l15) * F_IN;
    v8f acc[4];
    for (int i = 0; i < 4; ++i) acc[i] = vzero8();
    for (int k = 0; k < F_IN; k += 4) {
        v2f a;
        a.x = arow[k + hi * 2];
        a.y = arow[k + hi * 2 + 1];
        for (int ti = 0; ti < 4; ++ti) {
            int col = (wave * 4 + ti) * 16 + l15;
            int h = col >> 5, o = col & 31;
            const float* wb = w1 + (size_t)h * F_IN * O1 + o;
            v2f b;
            b.x = wb[(k + hi * 2) * O1];
            b.y = wb[(k + hi * 2 + 1) * O1];
            acc[ti] = wmma_f32_16x16x4(a, b, acc[ti]);
        }
    }
    for (int ti = 0; ti < 4; ++ti) {
        int col = (wave * 4 + ti) * 16 + l15;
        int h = col >> 5, o = col & 31;
        for (int r = 0; r < 8; ++r) {
            int row = mt * 16 + r + hi * 8;
            hp1[((size_t)h * N_P + row) * O1 + o] = acc[ti][r];
        }
    }
}

// s_src1/s_dst1[h][n] = sum_o tanh(hp1[h][n][o]) * a[h][o]
__global__ void k_gat1_scores(const float* __restrict__ hp1,
                              const float* __restrict__ asrc,
                              const float* __restrict__ adst,
                              float* __restrict__ ssrc, float* __restrict__ sdst) {
    int i = blockIdx.x * blockDim.x + threadIdx.x;   // 0..4095 = h*512+n
    if (i >= H1 * N_P) return;
    int h = i >> 9;
    const float* row = hp1 + (size_t)i * O1;
    float a = 0.f, d = 0.f;
    for (int o = 0; o < O1; ++o) {
        float t = tanhf(row[o]);
        a += t * asrc[h * O1 + o];
        d += t * adst[h * O1 + o];
    }
    ssrc[i] = a; sdst[i] = d;
}

// Attention + aggregation + bias + ELU for GAT1; writes hiddn_wp rows 3584..4095
// grid (512, 8), block 256
__global__ void k_gat1_attn(const float* __restrict__ hp1,
                            const float* __restrict__ ssrc,
                            const float* __restrict__ sdst,
                            const int* __restrict__ adj,
                            const float* __restrict__ b1,
                            float* __restrict__ hwp) {
    __shared__ float p[N_P];
    __shared__ float red[256];
    __shared__ int   am[N_P];
    int n = blockIdx.x, h = blockIdx.y, t = threadIdx.x;
    am[t]       = adj[n * N_P + t];
    am[t + 256] = adj[n * N_P + 256 + t];
    __syncthreads();
    float srcv = ssrc[h * N_P + n];
    float sc0 = srcv + sdst[h * N_P + t];
    sc0 = sc0 >= 0.f ? sc0 : NEG_SLOPE * sc0;
    float sc1 = srcv + sdst[h * N_P + 256 + t];
    sc1 = sc1 >= 0.f ? sc1 : NEG_SLOPE * sc1;
    bool m0 = am[t] > 0, m1 = am[t + 256] > 0;
    float mx = -INFINITY;
    if (m0) mx = sc0;
    if (m1 && sc1 > mx) mx = sc1;
    red[t] = mx; __syncthreads();
    for (int s = 128; s > 0; s >>= 1) {
        if (t < s && red[t + s] > red[t]) red[t] = red[t + s];
        __syncthreads();
    }
    mx = red[0]; __syncthreads();
    float e0 = m0 ? expf(sc0 - mx) : 0.f;
    float e1 = m1 ? expf(sc1 - mx) : 0.f;
    red[t] = e0 + e1; __syncthreads();
    for (int s = 128; s > 0; s >>= 1) {
        if (t < s) red[t] += red[t + s];
        __syncthreads();
    }
    float inv = red[0] > 0.f ? 1.f / red[0] : 0.f;
    __syncthreads();
    p[t] = e0 * inv; p[t + 256] = e1 * inv;
    __syncthreads();
    // out[o] = sum_m p[m] * hp1[h][m][o]
    int o = t & 31, ch = t >> 5;                    // 8 chunks of 64 m
    const float* hp = hp1 + (size_t)h * N_P * O1;
    float acc = 0.f;
    for (int m = ch * 64; m < ch * 64 + 64; ++m) acc += p[m] * hp[m * O1 + o];
    red[t] = acc; __syncthreads();
    if (ch == 0) {
        float s = 0.f;
        for (int c2 = 0; c2 < 8; ++c2) s += red[c2 * 32 + o];
        float v = s + b1[o];
        v = v > 0.f ? v : (expf(v) - 1.f);          // ELU
        hwp[(size_t)(N_W + n) * F_IN + h * O1 + o] = v;
    }
}

// ---------------------------------------------------------------------------
// GAT2 h_prime: hp2[h][n][o] = hiddn_wp[n,:] @ w2[h,:,o]  -- WMMA f32 16x16x4
// grid (256 mtiles, 8 heads), block 128 = 4 waves * (4 N-tiles each)
// ---------------------------------------------------------------------------
__global__ void k_gat2_hprime(const float* __restrict__ hwp,
                              const float* __restrict__ w2,
                              float* __restrict__ hp2) {
    int mt = blockIdx.x, h = blockIdx.y;
    int lane = threadIdx.x & 31, wave = threadIdx.x >> 5;
    int hi = lane >> 4, l15 = lane & 15;
    int row = mt * 16 + l15;
    const float* arow = hwp + (size_t)row * F_IN;
    const float* w2h  = w2 + (size_t)h * F_IN * O2;
    v8f acc[4];
    for (int i = 0; i < 4; ++i) acc[i] = vzero8();
    for (int k = 0; k < F_IN; k += 4) {
        v2f a;
        a.x = arow[k + hi * 2];
        a.y = arow[k + hi * 2 + 1];
        const float* bk = w2h + (size_t)(k + hi * 2) * O2;
        for (int ti = 0; ti < 4; ++ti) {
            int o0 = (wave * 4 + ti) * 16;
            v2f b;
            b.x = bk[o0 + l15];
            b.y = bk[O2 + o0 + l15];
            acc[ti] = wmma_f32_16x16x4(a, b, acc[ti]);
        }
    }
    for (int ti = 0; ti < 4; ++ti) {
        int o0 = (wave * 4 + ti) * 16;
        for (int r = 0; r < 8; ++r) {
            int rr = mt * 16 + r + hi * 8;
            hp2[((size_t)h * N_WP + rr) * O2 + o0 + l15] = acc[ti][r];
        }
    }
}

// s_src2/s_dst2[h][n] over O2=256
__global__ void k_gat2_scores(const float* __restrict__ hp2,
                              const float* __restrict__ asrc,
                              const float* __restrict__ adst,
                              float* __restrict__ ssrc, float* __restrict__ sdst) {
    int i = blockIdx.x * blockDim.x + threadIdx.x;   // h*4096+n
    if (i >= H2 * N_WP) return;
    int h = i >> 12;
    const float* row = hp2 + (size_t)i * O2;
    float a = 0.f, d = 0.f;
    for (int o = 0; o < O2; ++o) {
        float t = tanhf(row[o]);
        a += t * asrc[h * O2 + o];
        d += t * adst[h * O2 + o];
    }
    ssrc[i] = a; sdst[i] = d;
}

// Per (h,n): masked row max and 1/sum(exp) -- one adjacency-row pass, 8 heads
// grid 4096, block 256
__global__ void k_gat2_stats(const int* __restrict__ adj,
                             const float* __restrict__ ssrc,
                             const float* __restrict__ sdst,
                             float* __restrict__ rowmax, float* __restrict__ invden) {
    __shared__ int   am[N_WP];
    __shared__ float red[256];
    int n = blockIdx.x, t = threadIdx.x;
    for (int j = t; j < N_WP; j += 256) am[j] = adj[(size_t)n * N_WP + j];
    __syncthreads();
    for (int h = 0; h < H2; ++h) {
        float srcv = ssrc[h * N_WP + n];
        float mx = -INFINITY;
        for (int m = t; m < N_WP; m += 256) {
            if (am[m] > 0) {
                float x = srcv + sdst[h * N_WP + m];
                x = x >= 0.f ? x : NEG_SLOPE * x;
                if (x > mx) mx = x;
            }
        }
        red[t] = mx; __syncthreads();
        for (int s = 128; s > 0; s >>= 1) {
            if (t < s && red[t + s] > red[t]) red[t] = red[t + s];
            __syncthreads();
        }
        mx = red[0]; __syncthreads();
        float sum = 0.f;
        for (int m = t; m < N_WP; m += 256) {
            if (am[m] > 0) {
                float x = srcv + sdst[h * N_WP + m];
                x = x >= 0.f ? x : NEG_SLOPE * x;
                sum += expf(x - mx);
            }
        }
        red[t] = sum; __syncthreads();
        for (int s = 128; s > 0; s >>= 1) {
            if (t < s) red[t] += red[t + s];
            __syncthreads();
        }
        if (t == 0) {
            rowmax[h * N_WP + n] = mx;
            invden[h * N_WP + n] = red[0] > 0.f ? 1.f / red[0] : 0.f;
        }
        __syncthreads();
    }
}

// ---------------------------------------------------------------------------
// GAT2 aggregation (flash-style): hw[n][o] = (1/8) sum_h P_h[n,:] @ hp2[h] + b2
// adj tiles come in via async global->LDS (ASYNCcnt); softmax-normalized P
// tiles built in LDS purely from LDS-resident stats; WMMA accumulates across
// heads, fusing the head-mean + bias epilogue.
// grid (224 mtiles over output rows < 3584, 2 col-groups), block 128
// ---------------------------------------------------------------------------
#define KC 128
#define PS (KC + 4)   // padded LDS stride, avoids bank conflicts on A reads
__global__ void k_gat2_agg(const int* __restrict__ adj,
                           const float* __restrict__ hp2,
                           const float* __restrict__ ssrc,
                           const float* __restrict__ sdst,
                           const float* __restrict__ rowmax,
                           const float* __restrict__ invden,
                           const float* __restrict__ b2,
                           float* __restrict__ hw) {
    __shared__ __align__(16) int am[16 * KC];
    __shared__ float P[16 * PS];
    __shared__ float sD[H2][KC];              // sdst chunk, all heads
    __shared__ float rowS[H2 * 16], rowM[H2 * 16], rowI[H2 * 16];
    int mt = blockIdx.x, ng = blockIdx.y, t = threadIdx.x;
    int lane = t & 31, wave = t >> 5;
    int hi = lane >> 4, l15 = lane & 15;
    int o0 = ng * 128 + wave * 32;
    int o1 = o0 + 16;
    v8f acc0 = vzero8(), acc1 = vzero8();

    // per-row softmax stats for this M-tile, all heads (t == h*16+r exactly)
    {
        int h = t >> 4, r = t & 15;
        int n = mt * 16 + r;
        rowS[t] = ssrc[h * N_WP + n];
        rowM[t] = rowmax[h * N_WP + n];
        rowI[t] = invden[h * N_WP + n];
    }

    for (int kb = 0; kb < N_WP; kb += KC) {
        __syncthreads();   // previous iter done with am/P/sD
        // async-copy the 16xKC adjacency tile straight into LDS
        for (int g = t; g < (16 * KC) / 4; g += 128) {
            int e = g * 4;
            int r = e >> 7, c = e & (KC - 1);
            async_load_b128(adj + (size_t)(mt * 16 + r) * N_WP + kb + c, &am[e]);
        }
        // stage this chunk of s_dst for all heads
        for (int j = t; j < H2 * KC; j += 128) {
            int h = j >> 7, c = j & (KC - 1);
            sD[h][c] = sdst[h * N_WP + kb + c];
        }
        wait_async0();
        for (int h = 0; h < H2; ++h) {
            __syncthreads();   // tiles ready / previous wmma done with P
            for (int j = t; j < 16 * KC; j += 128) {
                int r = j >> 7, c = j & (KC - 1);
                float v = 0.f;
                if (am[j] > 0) {
                    float x = rowS[h * 16 + r] + sD[h][c];
                    x = x >= 0.f ? x : NEG_SLOPE * x;
                    v = expf(x - rowM[h * 16 + r]) * rowI[h * 16 + r];
                }
                P[r * PS + c] = v;
            }
            __syncthreads();
            const float* hp = hp2 + ((size_t)h * N_WP + kb) * O2;
            for (int kk = 0; kk < KC; kk += 4) {
                v2f a;
                a.x = P[l15 * PS + kk + hi * 2];
                a.y = P[l15 * PS + kk + hi * 2 + 1];
                const float* hpk = hp + (size_t)(kk + hi * 2) * O2;
                v2f b0, b1v;
                b0.x  = hpk[o0 + l15];
                b0.y  = hpk[O2 + o0 + l15];
                b1v.x = hpk[o1 + l15];
                b1v.y = hpk[O2 + o1 + l15];
                acc0 = wmma_f32_16x16x4(a, b0, acc0);
                acc1 = wmma_f32_16x16x4(a, b1v, acc1);
            }
        }
    }
    for (int r = 0; r < 8; ++r) {
        int row = mt * 16 + r + hi * 8;                 // < 3584 by construction
        hw[(size_t)row * O2 + o0 + l15] = acc0[r] * 0.125f + b2[o0 + l15];
        hw[(size_t)row * O2 + o1 + l15] = acc1[r] * 0.125f + b2[o1 + l15];
    }
}

// ---------------------------------------------------------------------------
// y = input(1024x3584) @ hiddn_w(3584x256)  -- WMMA
// grid (64 mtiles, 2 col-groups), block 128
// ---------------------------------------------------------------------------
__global__ void k_final_mm(const float* __restrict__ inp,
                           const float* __restrict__ hw,
                           float* __restrict__ y) {
    int mt = blockIdx.x, ng = blockIdx.y, t = threadIdx.x;
    int lane = t & 31, wave = t >> 5;
    int hi = lane >> 4, l15 = lane & 15;
    int o0 = ng * 128 + wave * 32, o1 = o0 + 16;
    const float* arow = inp + (size_t)(mt * 16 + l15) * N_W;
    v8f acc0 = vzero8(), acc1 = vzero8();
    for (int k = 0; k < N_W; k += 4) {
        v2f a;
        a.x = arow[k + hi * 2];
        a.y = arow[k + hi * 2 + 1];
        const float* hk = hw + (size_t)(k + hi * 2) * O2;
        v2f b0, b1v;
        b0.x  = hk[o0 + l15];
        b0.y  = hk[O2 + o0 + l15];
        b1v.x = hk[o1 + l15];
        b1v.y = hk[O2 + o1 + l15];
        acc0 = wmma_f32_16x16x4(a, b0, acc0);
        acc1 = wmma_f32_16x16x4(a, b1v, acc1);
    }
    for (int r = 0; r < 8; ++r) {
        int rr = mt * 16 + r + hi * 8;
        y[(size_t)rr * O2 + o0 + l15] = acc0[r];
        y[(size_t)rr * O2 + o1 + l15] = acc1[r];
    }
}

__global__ void k_rowsum(const float* __restrict__ inp, float* __restrict__ rs) {
    __shared__ float red[256];
    int b = blockIdx.x, t = threadIdx.x;
    float s = 0.f;
    for (int j = t; j < N_W; j += 256) s += inp[(size_t)b * N_W + j];
    red[t] = s; __syncthreads();
    for (int st = 128; st > 0; st >>= 1) {
        if (t < st) red[t] += red[t + st];
        __syncthreads();
    }
    if (t == 0) rs[b] = red[0];
}

// out[b][j] = (y[b,:] . W_fc[:,j]) / rowsum[b] + b_fc[j]
__global__ void k_fc(const float* __restrict__ y, const float* __restrict__ Wfc,
                     const float* __restrict__ bfc, const float* __restrict__ rs,
                     float* __restrict__ out) {
    __shared__ float red[5][256];
    int b = blockIdx.x, t = threadIdx.x;
    float yv = y[(size_t)b * O2 + t];
    for (int j = 0; j < 5; ++j) red[j][t] = yv * Wfc[t * 5 + j];
    __syncthreads();
    for (int st = 128; st > 0; st >>= 1) {
        if (t < st)
            for (int j = 0; j < 5; ++j) red[j][t] += red[j][t + st];
        __syncthreads();
    }
    if (t < 5) out[b * 5 + t] = red[t][0] / rs[b] + bfc[t];
}

// ---------------------------------------------------------------------------
extern "C" void kernel_launch(void* const* d_in, const int* in_sizes, int n_in,
                              void* d_out, int out_size, void* d_ws, size_t ws_size,
                              hipStream_t stream) {
    const float* input        = (const float*)d_in[0];
    const float* pern_feature = (const float*)d_in[1];
    const float* word_feature = (const float*)d_in[2];
    const float* w1           = (const float*)d_in[3];
    const float* a_src1       = (const float*)d_in[4];
    const float* a_dst1       = (const float*)d_in[5];
    const float* b1           = (const float*)d_in[6];
    const float* w2           = (const float*)d_in[7];
    const float* a_src2       = (const float*)d_in[8];
    const float* a_dst2       = (const float*)d_in[9];
    const float* b2           = (const float*)d_in[10];
    const float* W_fc         = (const float*)d_in[11];
    const float* b_fc         = (const float*)d_in[12];
    const int*   pern_adj     = (const int*)d_in[13];
    const int*   wp_adj       = (const int*)d_in[14];
    float* out = (float*)d_out;

    float* ws = (float*)d_ws;
    size_t off = 0;
    float* hp1   = ws + off; off += (size_t)H1 * N_P * O1;     // 131072
    float* ssrc1 = ws + off; off += H1 * N_P;
    float* sdst1 = ws + off; off += H1 * N_P;
    float* hwp   = ws + off; off += (size_t)N_WP * F_IN;       // 1M
    float* hp2   = ws + off; off += (size_t)H2 * N_WP * O2;    // 8.4M
    float* ssrc2 = ws + off; off += H2 * N_WP;
    float* sdst2 = ws + off; off += H2 * N_WP;
    float* rmax2 = ws + off; off += H2 * N_WP;
    float* iden2 = ws + off; off += H2 * N_WP;
    float* hw    = ws + off; off += (size_t)N_W * O2;          // 0.9M
    float* y     = ws + off; off += (size_t)BATCH * O2;
    float* rs    = ws + off; off += BATCH;

    // --- GAT1 ---
    k_gat1_hprime<<<N_P / 16, 128, 0, stream>>>(pern_feature, w1, hp1);
    k_gat1_scores<<<(H1 * N_P + 255) / 256, 256, 0, stream>>>(hp1, a_src1, a_dst1, ssrc1, sdst1);
    k_gat1_attn<<<dim3(N_P, H1), 256, 0, stream>>>(hp1, ssrc1, sdst1, pern_adj, b1, hwp);
    (void)hipMemcpyAsync(hwp, word_feature, (size_t)N_W * F_IN * sizeof(float),
                         hipMemcpyDeviceToDevice, stream);

    // --- GAT2 (WMMA + async-LDS) ---
    k_gat2_hprime<<<dim3(N_WP / 16, H2), 128, 0, stream>>>(hwp, w2, hp2);
    k_gat2_scores<<<(H2 * N_WP + 255) / 256, 256, 0, stream>>>(hp2, a_src2, a_dst2, ssrc2, sdst2);
    k_gat2_stats<<<N_WP, 256, 0, stream>>>(wp_adj, ssrc2, sdst2, rmax2, iden2);
    k_gat2_agg<<<dim3(N_W / 16, 2), 128, 0, stream>>>(wp_adj, hp2, ssrc2, sdst2,
                                                      rmax2, iden2, b2, hw);

    // --- readout (WMMA + small) ---
    k_final_mm<<<dim3(BATCH / 16, 2), 128, 0, stream>>>(input, hw, y);
    k_rowsum<<<BATCH, 256, 0, stream>>>(input, rs);
    k_fc<<<BATCH, 256, 0, stream>>>(y, W_fc, b_fc, rs, out);
}